// LightGCN_59854664237163
// MI455X (gfx1250) — compile-verified
//
#include <hip/hip_runtime.h>

#define NUM_USERS 100000
#define NUM_ITEMS 50000
#define NUM_NODES 150000   // NUM_USERS + NUM_ITEMS
#define DIM       64
#define NUM_EDGES 1200000
#define NUM_LAYERS 4

typedef __attribute__((ext_vector_type(2))) float v2f;
typedef __attribute__((ext_vector_type(8))) float v8f;

// ---------------- elementwise / setup kernels ----------------

__global__ void zero4_kernel(float4* __restrict__ p, int n4) {
    int i = blockIdx.x * blockDim.x + threadIdx.x;
    if (i < n4) p[i] = float4{0.f, 0.f, 0.f, 0.f};
}

__global__ void degree_kernel(const int* __restrict__ col, float* __restrict__ deg, int E) {
    int e = blockIdx.x * blockDim.x + threadIdx.x;
    if (e < E) atomicAdd(&deg[col[e]], 1.0f);
}

__global__ void rsqrt_kernel(float* __restrict__ deg, int n) {
    int i = blockIdx.x * blockDim.x + threadIdx.x;
    if (i < n) {
        float d = deg[i];
        deg[i] = (d > 0.f) ? rsqrtf(d) : 0.f;
    }
}

__global__ void norm_kernel(const int* __restrict__ row, const int* __restrict__ col,
                            const float* __restrict__ dinv, float* __restrict__ nrm, int E) {
    int e = blockIdx.x * blockDim.x + threadIdx.x;
    if (e < E) nrm[e] = dinv[row[e]] * dinv[col[e]];
}

// embA = concat(user,item); acc = same  (float4-vectorized)
__global__ void init_kernel(const float4* __restrict__ user, const float4* __restrict__ item,
                            float4* __restrict__ emb, float4* __restrict__ acc,
                            int n4user, int n4total) {
    int i = blockIdx.x * blockDim.x + threadIdx.x;
    if (i < n4total) {
        float4 v = (i < n4user) ? user[i] : item[i - n4user];
        emb[i] = v;
        acc[i] = v;
    }
}

__global__ void add4_kernel(float4* __restrict__ acc, const float4* __restrict__ x, int n4) {
    int i = blockIdx.x * blockDim.x + threadIdx.x;
    if (i < n4) {
        float4 a = acc[i], b = x[i];
        acc[i] = float4{a.x + b.x, a.y + b.y, a.z + b.z, a.w + b.w};
    }
}

// ---------------- edge-parallel SpMM: out[col] += norm * emb[row] ----------------
// 16 lanes per edge, one float4 (b128 gather) per lane, 4 float atomics per lane.
// Node table (38.4 MB) is L2-resident on MI455X (192 MB L2), so gathers/atomics hit L2.
__global__ void spmm_kernel(const int* __restrict__ row, const int* __restrict__ col,
                            const float* __restrict__ nrm,
                            const float4* __restrict__ embin, float* __restrict__ embout,
                            int E) {
    int t = blockIdx.x * blockDim.x + threadIdx.x;
    int e = t >> 4;
    int lane = t & 15;
    if (e >= E) return;
    int s = row[e];
    int d = col[e];
    float w = nrm[e];
    float4 v = embin[(size_t)s * 16 + lane];           // global_load_b128, L2 hit
    float* dst = embout + (size_t)d * DIM + lane * 4;
    atomicAdd(dst + 0, w * v.x);                        // global_atomic_add_f32
    atomicAdd(dst + 1, w * v.y);
    atomicAdd(dst + 2, w * v.z);
    atomicAdd(dst + 3, w * v.w);
}

// ---------------- finalize: acc *= 1/25 via V_WMMA_F32_16X16X4_F32 ----------------
// One wave per 16-node block. For each 16x16 column tile t, chain four K=4 WMMAs:
//   D = sum_q A_q x B_q,  A_q = acc[:, 16t+4q .. +4) (16x4 f32),
//   B_q[k][n] = s * (n == 4q+k)  -> D = s * acc_tile, exact in f32.
// ISA layouts (wave32):
//   A 16x4 : lanes 0-15 M=0..15 hold K={0,1} in VGPR{0,1}; lanes 16-31 hold K={2,3}.
//   B 4x16 : low lanes hold low-K rows: VGPR0 = row (hi?2:0), VGPR1 = row (hi?3:1), n=lane&15.
//   D 16x16: VGPR r, lanes 0-15 -> (M=r, N=lane), lanes 16-31 -> (M=r+8, N=lane-16).
__global__ void scale_wmma_kernel(float* __restrict__ out, int nblocks) {
    int wave = (int)((blockIdx.x * blockDim.x + threadIdx.x) >> 5);
    int lane = threadIdx.x & 31;
    if (wave >= nblocks) return;          // wave-uniform: EXEC stays all-ones for WMMA
    const float s = 1.0f / ((NUM_LAYERS + 1) * (NUM_LAYERS + 1));
    int lo = lane & 15;
    int hi = lane >> 4;
    float* base = out + (size_t)wave * 16 * DIM;   // 16 nodes x 64 features

    for (int t = 0; t < 4; ++t) {
        v8f c = {};
#pragma unroll
        for (int q = 0; q < 4; ++q) {
            int k0 = hi * 2;
            v2f a, b;
            a[0] = base[(size_t)lo * DIM + t * 16 + q * 4 + k0];
            a[1] = base[(size_t)lo * DIM + t * 16 + q * 4 + k0 + 1];
            b[0] = (lo == q * 4 + k0)     ? s : 0.0f;
            b[1] = (lo == q * 4 + k0 + 1) ? s : 0.0f;
            c = __builtin_amdgcn_wmma_f32_16x16x4_f32(
                    /*neg_a=*/false, a, /*neg_b=*/false, b,
                    /*c_mod=*/(short)0, c, /*reuse_a=*/false, /*reuse_b=*/false);
        }
#pragma unroll
        for (int r = 0; r < 8; ++r) {
            base[(size_t)(r + 8 * hi) * DIM + t * 16 + lo] = c[r];
        }
    }
}

// ---------------- host-side launcher ----------------

extern "C" void kernel_launch(void* const* d_in, const int* in_sizes, int n_in,
                              void* d_out, int out_size, void* d_ws, size_t ws_size,
                              hipStream_t stream) {
    const float* user = (const float*)d_in[0];
    const float* item = (const float*)d_in[1];
    const int*   ei   = (const int*)d_in[2];
    const int*   row  = ei;               // edge_index[0] = src
    const int*   col  = ei + NUM_EDGES;   // edge_index[1] = dst
    float* acc = (float*)d_out;           // [NUM_NODES, 64] = users then items

    // workspace carve-out (256B aligned): dinv | norm | embA | embB  (~82 MB)
    char*  ws  = (char*)d_ws;
    size_t off = 0;
    auto take = [&](size_t bytes) -> char* {
        char* p = ws + off;
        off += (bytes + 255) & ~(size_t)255;
        return p;
    };
    float* dinv = (float*)take((size_t)NUM_NODES * sizeof(float));
    float* nrm  = (float*)take((size_t)NUM_EDGES * sizeof(float));
    float* embA = (float*)take((size_t)NUM_NODES * DIM * sizeof(float));
    float* embB = (float*)take((size_t)NUM_NODES * DIM * sizeof(float));
    (void)ws_size; (void)in_sizes; (void)n_in; (void)out_size;

    const int B = 256;
    const int n4_nodes = NUM_NODES / 4;            // 37500 float4 (deg/dinv)
    const int n4_emb   = NUM_NODES * DIM / 4;      // 2,400,000 float4
    const int n4_user  = NUM_USERS * DIM / 4;      // 1,600,000 float4

    // 1) degree -> dinv
    zero4_kernel<<<(n4_nodes + B - 1) / B, B, 0, stream>>>((float4*)dinv, n4_nodes);
    degree_kernel<<<(NUM_EDGES + B - 1) / B, B, 0, stream>>>(col, dinv, NUM_EDGES);
    rsqrt_kernel<<<(NUM_NODES + B - 1) / B, B, 0, stream>>>(dinv, NUM_NODES);

    // 2) per-edge norm
    norm_kernel<<<(NUM_EDGES + B - 1) / B, B, 0, stream>>>(row, col, dinv, nrm, NUM_EDGES);

    // 3) emb0 and acc init
    init_kernel<<<(n4_emb + B - 1) / B, B, 0, stream>>>(
        (const float4*)user, (const float4*)item, (float4*)embA, (float4*)acc,
        n4_user, n4_emb);

    // 4) 4 propagation layers with ping-pong buffers
    float* cur = embA;
    float* nxt = embB;
    for (int l = 0; l < NUM_LAYERS; ++l) {
        zero4_kernel<<<(n4_emb + B - 1) / B, B, 0, stream>>>((float4*)nxt, n4_emb);
        spmm_kernel<<<((NUM_EDGES * 16) + B - 1) / B, B, 0, stream>>>(
            row, col, nrm, (const float4*)cur, nxt, NUM_EDGES);
        add4_kernel<<<(n4_emb + B - 1) / B, B, 0, stream>>>(
            (float4*)acc, (const float4*)nxt, n4_emb);
        float* tmp = cur; cur = nxt; nxt = tmp;
    }

    // 5) acc *= 1/25 via fp32 WMMA (exact)
    const int nblocks = NUM_NODES / 16;            // 9375 waves
    const int waves_per_block = B / 32;
    scale_wmma_kernel<<<(nblocks + waves_per_block - 1) / waves_per_block, B, 0, stream>>>(
        acc, nblocks);
}